// TSP_Decoder_7069516169268
// MI455X (gfx1250) — compile-verified
//
#include <hip/hip_runtime.h>
#include <hip/hip_bf16.h>
#include <math.h>

// ---------------------------------------------------------------------------
// TSP decoder forward, MI455X (gfx1250), wave32, V_WMMA_F32_16X16X4_F32 path.
// B=16, N=M=1000, E=128, H=8, D=16.
// ---------------------------------------------------------------------------

typedef __attribute__((ext_vector_type(2))) float v2f;
typedef __attribute__((ext_vector_type(8))) float v8f;

#define EDIM 128
#define NN   1000
#define NT   63          // ceil(1000/16)
#define MPAD (NT * 16)   // 1008

__device__ __forceinline__ v8f wmma4(v2f a, v2f b, v8f c) {
  // D = A(16x4,f32) * B(4x16,f32) + C(16x16,f32)
  return __builtin_amdgcn_wmma_f32_16x16x4_f32(
      /*neg_a=*/false, a, /*neg_b=*/false, b,
      /*c_mod=*/(short)0, c, /*reuse_a=*/false, /*reuse_b=*/false);
}

// ---------------------------------------------------------------------------
// Kernel 1: projections.  grid = (1000 row-tiles, 3)   block = 256 (8 waves)
//   y==0: Kmat = X @ Wk      y==1: Vmat = X @ Wv
//   y==2: Qmat = Q1 @ Wq_first + XL @ Wq_last
// Wave j computes output columns [16j, 16j+16).
// ---------------------------------------------------------------------------
__global__ __launch_bounds__(256) void k_proj(
    const float* __restrict__ X, const float* __restrict__ Q1,
    const float* __restrict__ XL,
    const float* __restrict__ Wk, const float* __restrict__ Wv,
    const float* __restrict__ Wqf, const float* __restrict__ Wql,
    float* __restrict__ Kw, float* __restrict__ Vw, float* __restrict__ Qw) {
  const int tile = blockIdx.x;           // rows of flat [16000,128]
  const int which = blockIdx.y;
  const int wave = threadIdx.x >> 5;
  const int lane = threadIdx.x & 31;
  const int half = lane >> 4;            // 0: lanes 0-15, 1: lanes 16-31
  const int l16 = lane & 15;
  const int rowbase = tile * 16;
  const int col = wave * 16 + l16;

  const float* A0;
  const float* W0;
  const float* A1 = nullptr;
  const float* W1 = nullptr;
  float* Dst;
  if (which == 0) { A0 = X;  W0 = Wk;  Dst = Kw; }
  else if (which == 1) { A0 = X;  W0 = Wv;  Dst = Vw; }
  else { A0 = Q1; W0 = Wqf; A1 = XL; W1 = Wql; Dst = Qw; }

  v8f acc = {};
  {
    const float* arow = A0 + (size_t)(rowbase + l16) * EDIM;
    for (int k = 0; k < EDIM; k += 4) {
      v2f a, b;
      a.x = arow[k + 2 * half];
      a.y = arow[k + 2 * half + 1];
      b.x = W0[(k + 2 * half) * EDIM + col];
      b.y = W0[(k + 2 * half + 1) * EDIM + col];
      acc = wmma4(a, b, acc);
    }
  }
  if (A1) {
    const float* arow = A1 + (size_t)(rowbase + l16) * EDIM;
    for (int k = 0; k < EDIM; k += 4) {
      v2f a, b;
      a.x = arow[k + 2 * half];
      a.y = arow[k + 2 * half + 1];
      b.x = W1[(k + 2 * half) * EDIM + col];
      b.y = W1[(k + 2 * half + 1) * EDIM + col];
      acc = wmma4(a, b, acc);
    }
  }
#pragma unroll
  for (int g = 0; g < 8; ++g) {
    int r = rowbase + g + 8 * half;      // C/D layout: rows g / g+8
    Dst[(size_t)r * EDIM + col] = acc[g];
  }
}

// ---------------------------------------------------------------------------
// Kernel 2: attention (online softmax) + multi_head_combine, fused.
// grid = 16 * 63 (batch x 16-row n-tile), block = 256; wave h == head h.
// ---------------------------------------------------------------------------
__global__ __launch_bounds__(256) void k_attn(
    const float* __restrict__ Qw, const float* __restrict__ Kw,
    const float* __restrict__ Vw, const float* __restrict__ Msk,
    const float* __restrict__ Wc, const float* __restrict__ bc,
    float* __restrict__ OUT) {
  __shared__ float lds_p[8][16 * 16];    // per-wave P-tile transpose buffer
  __shared__ float lds_ctx[16 * EDIM];   // assembled [16,128] context tile

  const int b = blockIdx.x / NT;
  const int tile = blockIdx.x % NT;
  const int nbase = tile * 16;
  const int h = threadIdx.x >> 5;
  const int lane = threadIdx.x & 31;
  const int half = lane >> 4;
  const int l16 = lane & 15;
  const int hd = h * 16;

  const size_t bOff = (size_t)b * NN * EDIM;
  const float* Qb = Qw + bOff;
  const float* Kb = Kw + bOff;
  const float* Vb = Vw + bOff;
  const float* Mb = Msk + (size_t)b * NN * NN;

  int nrowA = nbase + l16; if (nrowA > NN - 1) nrowA = NN - 1;

  // Q tile as WMMA A fragments (16 rows x 16 head-dims, chunks of K=4)
  v2f aq[4];
#pragma unroll
  for (int kc = 0; kc < 4; ++kc) {
    aq[kc].x = Qb[(size_t)nrowA * EDIM + hd + 4 * kc + 2 * half];
    aq[kc].y = Qb[(size_t)nrowA * EDIM + hd + 4 * kc + 2 * half + 1];
  }

  v8f acc = {};
  float mrun[8], lrun[8];
#pragma unroll
  for (int g = 0; g < 8; ++g) { mrun[g] = -1e30f; lrun[g] = 0.0f; }

  float* pmem = lds_p[h];
  const int mcol = 0;
  (void)mcol;

  for (int mt = 0; mt < NT; ++mt) {
    const int mbase = mt * 16;
    int mc = mbase + l16;
    int mcl = mc > NN - 1 ? NN - 1 : mc;

    // S = Q . K^T  (B operand = K transposed: element (d, m) = K[m][d])
    v8f S = {};
#pragma unroll
    for (int kc = 0; kc < 4; ++kc) {
      v2f bk;
      bk.x = Kb[(size_t)mcl * EDIM + hd + 4 * kc + 2 * half];
      bk.y = Kb[(size_t)mcl * EDIM + hd + 4 * kc + 2 * half + 1];
      S = wmma4(aq[kc], bk, S);
    }

    // mask + online-softmax update (row stats live replicated per 16-lane half)
#pragma unroll
    for (int g = 0; g < 8; ++g) {
      int r = nbase + g + 8 * half; if (r > NN - 1) r = NN - 1;
      float mv = (mc < NN) ? Mb[(size_t)r * NN + mc] : -1e30f;
      float s = S[g] * 0.25f + mv;               // 1/sqrt(16)
      float rmax = s;
      for (int off = 8; off >= 1; off >>= 1)
        rmax = fmaxf(rmax, __shfl_xor(rmax, off));
      float newm = fmaxf(mrun[g], rmax);
      float alpha = __expf(mrun[g] - newm);
      float pv = __expf(s - newm);
      float rsum = pv;
      for (int off = 8; off >= 1; off >>= 1)
        rsum += __shfl_xor(rsum, off);
      lrun[g] = lrun[g] * alpha + rsum;
      mrun[g] = newm;
      acc[g] *= alpha;
      pmem[(g + 8 * half) * 16 + l16] = pv;      // C-layout -> LDS
    }
    // wave-private LDS transpose: wait for DS stores, block compiler motion
    asm volatile("s_wait_dscnt 0x0" ::: "memory");

    // acc += P @ V   (A = P rows n x K=m, B = V rows m x cols d)
#pragma unroll
    for (int kc = 0; kc < 4; ++kc) {
      v2f ap, bv;
      ap.x = pmem[l16 * 16 + 4 * kc + 2 * half];
      ap.y = pmem[l16 * 16 + 4 * kc + 2 * half + 1];
      int m0 = mbase + 4 * kc + 2 * half;
      int m1 = m0 + 1;
      if (m0 > NN - 1) m0 = NN - 1;
      if (m1 > NN - 1) m1 = NN - 1;
      bv.x = Vb[(size_t)m0 * EDIM + hd + l16];
      bv.y = Vb[(size_t)m1 * EDIM + hd + l16];
      acc = wmma4(ap, bv, acc);
    }
  }

  // normalize, assemble per-head context into shared [16,128] tile
#pragma unroll
  for (int g = 0; g < 8; ++g) {
    float inv = 1.0f / lrun[g];
    lds_ctx[(g + 8 * half) * EDIM + hd + l16] = acc[g] * inv;
  }
  __syncthreads();

  // fused combine: out_tile = ctx @ Wc + bc ; wave h -> cols [hd, hd+16)
  v8f oc = {};
  for (int k = 0; k < EDIM; k += 4) {
    v2f a, b;
    a.x = lds_ctx[l16 * EDIM + k + 2 * half];
    a.y = lds_ctx[l16 * EDIM + k + 2 * half + 1];
    b.x = Wc[(k + 2 * half) * EDIM + hd + l16];
    b.y = Wc[(k + 2 * half + 1) * EDIM + hd + l16];
    oc = wmma4(a, b, oc);
  }
  float bias = bc[hd + l16];
#pragma unroll
  for (int g = 0; g < 8; ++g) {
    int r = nbase + g + 8 * half;
    if (r < NN) OUT[bOff + (size_t)r * EDIM + hd + l16] = oc[g] + bias;
  }
}

// ---------------------------------------------------------------------------
// Kernel 3: pointer scores + clipped-tanh + masked softmax over M=1000.
// grid = 16 * 63, block = 256. Logits for the 16-row tile staged in LDS.
// ---------------------------------------------------------------------------
__global__ __launch_bounds__(256) void k_ptr(
    const float* __restrict__ OUT, const float* __restrict__ nodes,
    const float* __restrict__ Msk, float* __restrict__ out) {
  __shared__ float lds_logit[16 * MPAD];   // 64512 bytes

  const int b = blockIdx.x / NT;
  const int tile = blockIdx.x % NT;
  const int nbase = tile * 16;
  const int h = threadIdx.x >> 5;
  const int lane = threadIdx.x & 31;
  const int half = lane >> 4;
  const int l16 = lane & 15;

  const float* Ob = OUT + (size_t)b * NN * EDIM;
  const float* Xb = nodes + (size_t)b * NN * EDIM;
  const float* Mb = Msk + (size_t)b * NN * NN;

  int nrowA = nbase + l16; if (nrowA > NN - 1) nrowA = NN - 1;
  const float* orow = Ob + (size_t)nrowA * EDIM;

  // each wave takes m-tiles h, h+8, ... : logits = out . nodes^T
  for (int mt = h; mt < NT; mt += 8) {
    const int mbase = mt * 16;
    int mc = mbase + l16;
    int mcl = mc > NN - 1 ? NN - 1 : mc;
    const float* xrow = Xb + (size_t)mcl * EDIM;   // B elem (e,m)=nodes[m][e]
    v8f S = {};
    for (int k = 0; k < EDIM; k += 4) {
      v2f a, bb;
      a.x = orow[k + 2 * half];
      a.y = orow[k + 2 * half + 1];
      bb.x = xrow[k + 2 * half];
      bb.y = xrow[k + 2 * half + 1];
      S = wmma4(a, bb, S);
    }
#pragma unroll
    for (int g = 0; g < 8; ++g) {
      int rloc = g + 8 * half;
      int r = nbase + rloc; int rc = r > NN - 1 ? NN - 1 : r;
      float t;
      if (mc < NN)
        t = 10.0f * tanhf(S[g] * 0.08838834764831845f)  // 1/sqrt(128)
            + Mb[(size_t)rc * NN + mc];
      else
        t = -1e30f;
      lds_logit[rloc * MPAD + mbase + l16] = t;
    }
  }
  __syncthreads();

  // softmax per row; wave h owns rows h and h+8
#pragma unroll
  for (int rr = 0; rr < 2; ++rr) {
    const int rloc = h + 8 * rr;
    const int r = nbase + rloc;
    float* row = lds_logit + rloc * MPAD;
    float mx = -1e30f;
    for (int m = lane; m < MPAD; m += 32) mx = fmaxf(mx, row[m]);
    for (int off = 16; off >= 1; off >>= 1)
      mx = fmaxf(mx, __shfl_xor(mx, off));
    float sum = 0.0f;
    for (int m = lane; m < MPAD; m += 32) {
      float e = __expf(row[m] - mx);
      row[m] = e;
      sum += e;
    }
    for (int off = 16; off >= 1; off >>= 1) sum += __shfl_xor(sum, off);
    float inv = 1.0f / sum;
    if (r < NN) {
      float* dst = out + (size_t)b * NN * NN + (size_t)r * NN;
      for (int m = lane; m < NN; m += 32) dst[m] = row[m] * inv;
    }
  }
}

// ---------------------------------------------------------------------------
extern "C" void kernel_launch(void* const* d_in, const int* in_sizes, int n_in,
                              void* d_out, int out_size, void* d_ws,
                              size_t ws_size, hipStream_t stream) {
  (void)in_sizes; (void)n_in; (void)out_size; (void)ws_size;
  const float* X   = (const float*)d_in[0];   // encoded_nodes   [16,1000,128]
  const float* Q1  = (const float*)d_in[1];   // encoded_q1
  const float* XL  = (const float*)d_in[2];   // encoded_last_node
  const float* Msk = (const float*)d_in[3];   // ninf_mask       [16,1000,1000]
  const float* Wqf = (const float*)d_in[4];
  const float* Wql = (const float*)d_in[5];
  const float* Wk  = (const float*)d_in[6];
  const float* Wv  = (const float*)d_in[7];
  const float* Wc  = (const float*)d_in[8];
  const float* bcp = (const float*)d_in[9];
  float* outp = (float*)d_out;

  const size_t P = (size_t)16 * NN * EDIM;    // 2,048,000 floats
  float* Qw   = (float*)d_ws;                 // 4 x 8.192 MB scratch
  float* Kw   = Qw + P;
  float* Vw   = Kw + P;
  float* OUTw = Vw + P;

  dim3 blk(256);
  k_proj<<<dim3(1000, 3), blk, 0, stream>>>(X, Q1, XL, Wk, Wv, Wqf, Wql,
                                            Kw, Vw, Qw);
  k_attn<<<dim3(16 * NT), blk, 0, stream>>>(Qw, Kw, Vw, Msk, Wc, bcp, OUTw);
  k_ptr<<<dim3(16 * NT), blk, 0, stream>>>(OUTw, X, Msk, outp);
}